// FilterByOctaves_29661044146869
// MI455X (gfx1250) — compile-verified
//
#include <hip/hip_runtime.h>
#include <hip/hip_bf16.h>

// Problem constants (from the reference)
#define B_    256
#define T_    48000
#define NSEC  3
// Chunked linear-recurrence decomposition
#define L_    480      // chunk length (multiple of 16)
#define K_    100      // real chunks per row, L_*K_ == T_
#define KT_   112      // padded chunk count (7 WMMA M-tiles of 16)
#define MT_   7        // M tiles per row
#define NT_   30       // N tiles per chunk (L_/16)
#define LROW_ (KT_ * L_)   // padded ybuf row stride = 53760

typedef float v2f __attribute__((ext_vector_type(2)));
typedef float v4f __attribute__((ext_vector_type(4)));
typedef float v8f __attribute__((ext_vector_type(8)));

// ---- one step of the cascaded DF2T biquad, linear in (state, x) ----
__device__ inline void sos_step(const float* cf, const float* z, float x,
                                float* nz, float& yout) {
    float v = x;
#pragma unroll
    for (int s = 0; s < NSEC; ++s) {
        float b0 = cf[s * 5 + 0], b1 = cf[s * 5 + 1], b2 = cf[s * 5 + 2];
        float a1 = cf[s * 5 + 3], a2 = cf[s * 5 + 4];
        float y = fmaf(b0, v, z[2 * s]);
        nz[2 * s]     = fmaf(b1, v, fmaf(-a1, y, z[2 * s + 1]));
        nz[2 * s + 1] = fmaf(b2, v, -a2 * y);
        v = y;
    }
    yout = v;
}

// ---- Phase 0: derive A (6x6), c (6); precompute G[t]=c*A^t and A^L; zero row maxes ----
__global__ void iir_phase0(const float* __restrict__ sos, float* __restrict__ G,
                           float* __restrict__ AL, unsigned* __restrict__ rm1,
                           unsigned* __restrict__ rm2) {
    int tid = threadIdx.x;
    if (tid < B_) { rm1[tid] = 0u; rm2[tid] = 0u; }
    if (tid != 0 || blockIdx.x != 0) return;

    float cf[NSEC * 5];
#pragma unroll
    for (int s = 0; s < NSEC; ++s) {
        cf[s * 5 + 0] = sos[s * 6 + 0];
        cf[s * 5 + 1] = sos[s * 6 + 1];
        cf[s * 5 + 2] = sos[s * 6 + 2];
        cf[s * 5 + 3] = sos[s * 6 + 4];  // a1
        cf[s * 5 + 4] = sos[s * 6 + 5];  // a2
    }
    float A[36], c[6];
    for (int j = 0; j < 6; ++j) {
        float z[6] = {0, 0, 0, 0, 0, 0};
        z[j] = 1.0f;
        float nz[6], y;
        sos_step(cf, z, 0.0f, nz, y);
        for (int i = 0; i < 6; ++i) A[i * 6 + j] = nz[i];
        c[j] = y;
    }
    // G[t] = c * A^t  (row vector), padded to 8
    float g[6];
    for (int i = 0; i < 6; ++i) g[i] = c[i];
    for (int t = 0; t < L_; ++t) {
        for (int i = 0; i < 6; ++i) G[t * 8 + i] = g[i];
        G[t * 8 + 6] = 0.0f; G[t * 8 + 7] = 0.0f;
        float ng[6];
        for (int i = 0; i < 6; ++i) {           // g' = A^T g
            float acc = 0.0f;
            for (int j = 0; j < 6; ++j) acc = fmaf(A[j * 6 + i], g[j], acc);
            ng[i] = acc;
        }
        for (int i = 0; i < 6; ++i) g[i] = ng[i];
    }
    // AL = A^L
    float M[36];
    for (int i = 0; i < 36; ++i) M[i] = (i % 7 == 0) ? 1.0f : 0.0f;
    for (int t = 0; t < L_; ++t) {
        float NM[36];
        for (int i = 0; i < 6; ++i)
            for (int k = 0; k < 6; ++k) {
                float acc = 0.0f;
                for (int j = 0; j < 6; ++j) acc = fmaf(A[i * 6 + j], M[j * 6 + k], acc);
                NM[i * 6 + k] = acc;
            }
        for (int i = 0; i < 36; ++i) M[i] = NM[i];
    }
    for (int i = 0; i < 36; ++i) AL[i] = M[i];
}

// ---- Phase 1: per-chunk zero-state response + final partial state (float4 I/O) ----
__global__ __launch_bounds__(256) void iir_phase1(const float* __restrict__ x,
                                                  const float* __restrict__ sos,
                                                  float* __restrict__ ybuf,
                                                  float* __restrict__ P) {
    int id = blockIdx.x * blockDim.x + threadIdx.x;
    if (id >= B_ * K_) return;
    int row = id / K_, k = id % K_;
    float b0[NSEC], b1c[NSEC], b2c[NSEC], a1c[NSEC], a2c[NSEC];
#pragma unroll
    for (int s = 0; s < NSEC; ++s) {
        b0[s] = sos[s * 6 + 0]; b1c[s] = sos[s * 6 + 1]; b2c[s] = sos[s * 6 + 2];
        a1c[s] = sos[s * 6 + 4]; a2c[s] = sos[s * 6 + 5];
    }
    float z[6] = {0, 0, 0, 0, 0, 0};
    const float* xp = x + (size_t)row * T_ + (size_t)k * L_;
    float* yp = ybuf + (size_t)row * LROW_ + (size_t)k * L_;
    for (int tb = 0; tb < L_; tb += 16) {
        __builtin_prefetch(xp + tb + 64, 0, 3);
#pragma unroll
        for (int u = 0; u < 4; ++u) {
            int t = tb + u * 4;
            v4f xv = *(const v4f*)(xp + t);
            v4f yo;
#pragma unroll
            for (int e = 0; e < 4; ++e) {
                float v = xv[e];
#pragma unroll
                for (int s = 0; s < NSEC; ++s) {
                    float y = fmaf(b0[s], v, z[2 * s]);
                    z[2 * s]     = fmaf(b1c[s], v, fmaf(-a1c[s], y, z[2 * s + 1]));
                    z[2 * s + 1] = fmaf(b2c[s], v, -a2c[s] * y);
                    v = y;
                }
                yo[e] = v;
            }
            *(v4f*)(yp + t) = yo;
        }
    }
#pragma unroll
    for (int i = 0; i < 6; ++i) P[(size_t)id * 8 + i] = z[i];
}

// ---- Phase 2: per-row scan over chunks: s_{k+1} = A^L s_k + p_k ; zero pad rows ----
__global__ __launch_bounds__(256) void iir_phase2(const float* __restrict__ P,
                                                  const float* __restrict__ AL,
                                                  float* __restrict__ Sinit) {
    int b = blockIdx.x * blockDim.x + threadIdx.x;
    if (b >= B_) return;
    float M[36];
#pragma unroll
    for (int i = 0; i < 36; ++i) M[i] = AL[i];
    float s[6] = {0, 0, 0, 0, 0, 0};
    for (int k = 0; k < K_; ++k) {
        float* dst = Sinit + ((size_t)b * KT_ + k) * 8;
#pragma unroll
        for (int i = 0; i < 6; ++i) dst[i] = s[i];
        dst[6] = 0.0f; dst[7] = 0.0f;
        const float* p = P + ((size_t)b * K_ + k) * 8;
        float ns[6];
#pragma unroll
        for (int i = 0; i < 6; ++i) {
            float acc = p[i];
#pragma unroll
            for (int j = 0; j < 6; ++j) acc = fmaf(M[i * 6 + j], s[j], acc);
            ns[i] = acc;
        }
#pragma unroll
        for (int i = 0; i < 6; ++i) s[i] = ns[i];
    }
    // zero padded chunk rows so phantom WMMA output rows are exactly 0
    for (int k = K_; k < KT_; ++k) {
        float* dst = Sinit + ((size_t)b * KT_ + k) * 8;
#pragma unroll
        for (int i = 0; i < 8; ++i) dst[i] = 0.0f;
    }
}

// ---- Phase 3: WMMA rank-6 state correction  D[chunk][t] = S[chunk][:]*G[t][:]^T ----
// One wave per (row, 16-chunk tile); two chained V_WMMA_F32_16X16X4_F32 per 16-offset tile.
// All stores unconditional (padded row stride); max-abs masked with branch-free select.
__global__ __launch_bounds__(32) void iir_phase3(const float* __restrict__ G,
                                                 const float* __restrict__ Sinit,
                                                 float* __restrict__ ybuf,
                                                 unsigned* __restrict__ rm1) {
    int b  = blockIdx.x;        // row
    int mt = blockIdx.y;        // chunk tile
    int l  = threadIdx.x;       // lane 0..31
    int cb = mt * 16;
    int lo = l & 15;
    int hi = l >> 4;            // 0 or 1
    // A fragment: 16x4 f32 -> VGPR0 holds K=0 (lanes 0-15) / K=2 (lanes 16-31); VGPR1: K=1/K=3
    const float* srow = Sinit + ((size_t)b * KT_ + cb + lo) * 8;
    v2f a0 = *(const v2f*)(srow + 0 + 2 * hi);   // states 0..3
    v2f a1 = *(const v2f*)(srow + 4 + 2 * hi);   // states 4..7 (6,7 == 0)
    // D VGPR r maps to chunk m = cb + r + 8*hi, column N = lo
    float* ybase = ybuf + (size_t)b * LROW_ + (size_t)(cb + 8 * hi) * L_ + lo;
    float localmax = 0.0f;
    for (int nt = 0; nt < NT_; ++nt) {
        int ob = nt * 16;
        const float* grow = G + (size_t)(ob + lo) * 8;
        v2f g0 = *(const v2f*)(grow + 0 + 2 * hi);
        v2f g1 = *(const v2f*)(grow + 4 + 2 * hi);
        float* yp = ybase + ob;
        float yv[8];
#pragma unroll
        for (int r = 0; r < 8; ++r) yv[r] = yp[(size_t)r * L_];
        v8f acc = {};
        acc = __builtin_amdgcn_wmma_f32_16x16x4_f32(false, a0, false, g0,
                                                    (short)0, acc, false, false);
        acc = __builtin_amdgcn_wmma_f32_16x16x4_f32(false, a1, false, g1,
                                                    (short)0, acc, false, false);
#pragma unroll
        for (int r = 0; r < 8; ++r) {
            float y = yv[r] + acc[r];
            yp[(size_t)r * L_] = y;
            bool valid = (cb + 8 * hi + r) < K_;
            localmax = fmaxf(localmax, valid ? fabsf(y) : 0.0f);
        }
    }
#pragma unroll
    for (int off = 16; off > 0; off >>= 1)
        localmax = fmaxf(localmax, __shfl_xor(localmax, off, 32));
    if (l == 0) atomicMax(rm1 + b, __float_as_uint(localmax));
}

// ---- Phase 4: data = 0.9 x + 0.1 y / max|y| ; track per-row max|data| (float4) ----
__global__ __launch_bounds__(256) void iir_phase4(const float* __restrict__ x,
                                                  const float* __restrict__ ybuf,
                                                  const unsigned* __restrict__ rm1,
                                                  float* __restrict__ out,
                                                  unsigned* __restrict__ rm2) {
    int b = blockIdx.y;
    float inv = 0.1f / __uint_as_float(rm1[b]);
    float lm = 0.0f;
    int t = blockIdx.x * 1024 + threadIdx.x * 4;
    if (t < T_) {                         // T_ % 4 == 0, so full float4 is in range
        v4f xv = *(const v4f*)(x + (size_t)b * T_ + t);
        v4f yv = *(const v4f*)(ybuf + (size_t)b * LROW_ + t);
        v4f d;
#pragma unroll
        for (int e = 0; e < 4; ++e) {
            d[e] = fmaf(0.9f, xv[e], yv[e] * inv);
            lm = fmaxf(lm, fabsf(d[e]));
        }
        *(v4f*)(out + (size_t)b * T_ + t) = d;
    }
    __shared__ float red[8];
#pragma unroll
    for (int off = 16; off > 0; off >>= 1)
        lm = fmaxf(lm, __shfl_xor(lm, off, 32));
    if ((threadIdx.x & 31) == 0) red[threadIdx.x >> 5] = lm;
    __syncthreads();
    if (threadIdx.x == 0) {
        float m = red[0];
#pragma unroll
        for (int w = 1; w < 8; ++w) m = fmaxf(m, red[w]);
        atomicMax(rm2 + b, __float_as_uint(m));
    }
}

// ---- Phase 5: final normalization in place (float4) ----
__global__ __launch_bounds__(256) void iir_phase5(float* __restrict__ out,
                                                  const unsigned* __restrict__ rm2) {
    size_t id4 = (size_t)blockIdx.x * blockDim.x + threadIdx.x;
    size_t t = id4 * 4;
    if (t < (size_t)B_ * T_) {            // rows are 4-aligned, vectors never straddle
        int b = (int)(t / T_);
        float inv = 1.0f / __uint_as_float(rm2[b]);
        v4f d = *(const v4f*)(out + t);
#pragma unroll
        for (int e = 0; e < 4; ++e) d[e] *= inv;
        *(v4f*)(out + t) = d;
    }
}

extern "C" void kernel_launch(void* const* d_in, const int* in_sizes, int n_in,
                              void* d_out, int out_size, void* d_ws, size_t ws_size,
                              hipStream_t stream) {
    (void)in_sizes; (void)n_in; (void)out_size; (void)ws_size;
    const float* x   = (const float*)d_in[0];   // [256, 48000]
    const float* sos = (const float*)d_in[1];   // [3, 6]
    float* out = (float*)d_out;

    float* ws    = (float*)d_ws;
    float* ybuf  = ws;                                  // B * LROW_ (padded rows)
    float* P     = ybuf  + (size_t)B_ * LROW_;          // B*K*8
    float* Sinit = P     + (size_t)B_ * K_ * 8;         // B*KT*8
    float* G     = Sinit + (size_t)B_ * KT_ * 8;        // L*8
    float* AL    = G     + (size_t)L_ * 8;              // 64 (36 used)
    unsigned* rm1 = (unsigned*)(AL + 64);               // 256
    unsigned* rm2 = rm1 + B_;                           // 256

    iir_phase0<<<1, 256, 0, stream>>>(sos, G, AL, rm1, rm2);
    iir_phase1<<<(B_ * K_ + 255) / 256, 256, 0, stream>>>(x, sos, ybuf, P);
    iir_phase2<<<1, 256, 0, stream>>>(P, AL, Sinit);
    iir_phase3<<<dim3(B_, MT_), 32, 0, stream>>>(G, Sinit, ybuf, rm1);
    iir_phase4<<<dim3((T_ + 1023) / 1024, B_), 256, 0, stream>>>(x, ybuf, rm1, out, rm2);
    iir_phase5<<<(int)(((size_t)B_ * T_ / 4 + 255) / 256), 256, 0, stream>>>(out, rm2);
}